// Step_3_categories_56040733278864
// MI455X (gfx1250) — compile-verified
//
#include <hip/hip_runtime.h>

typedef __attribute__((ext_vector_type(2))) float v2f;
typedef __attribute__((ext_vector_type(8))) float v8f;

#define NUM_SPANS 512
#define NUM_PAIRS 100000
#define HIDDEN 768
#define NUM_CATS 13
#define P_STRIDE 32   // padded N (26 valid cols: 0..12 aspect, 13..25 opinion)

// -----------------------------------------------------------------------------
// Stage 1: P[s][n] = dot(emb[s, :768], Wcat[n, :768])
//   Wcat[n][k] = (n < 13) ? W[n][k] : W[n-13][768+k],  zero for n >= 26
// One wave (32 lanes) computes one 16x16 tile of P via V_WMMA_F32_16X16X4_F32.
// Grid: 32 M-tiles x 2 N-tiles = 64 waves.
// -----------------------------------------------------------------------------
__global__ __launch_bounds__(32) void partial_logits_wmma(
    const float* __restrict__ emb,   // [512*768]
    const float* __restrict__ W,     // [13*1536]
    float* __restrict__ P)           // [512*32] workspace
{
    const int tile  = blockIdx.x;     // 0..63
    const int mtile = tile >> 1;      // 0..31
    const int ntile = tile & 1;       // 0..1
    const int lane  = threadIdx.x;    // 0..31
    const int lm    = lane & 15;
    const int khalf = (lane >> 4) << 1;   // lanes 0-15 -> K pair {0,1}; 16-31 -> {2,3}

    // A fragment source: row of emb
    const int arow = mtile * 16 + lm;                 // 0..511
    const float* __restrict__ aptr = emb + arow * HIDDEN + khalf;

    // B fragment source: column n of Wcat (clamped + masked for n>=26)
    const int ncol   = ntile * 16 + lm;               // 0..31
    const bool valid = (ncol < 26);
    const int  wrow  = valid ? ((ncol < 13) ? ncol : (ncol - 13)) : 0;
    const int  woff  = (ncol >= 13 && valid) ? HIDDEN : 0;
    const float msk  = valid ? 1.0f : 0.0f;
    const float* __restrict__ bptr = W + wrow * (2 * HIDDEN) + woff + khalf;

    v8f acc0 = {};
    v8f acc1 = {};

    // K loop: 768 / 4 = 192 WMMAs, 2-way interleaved to break the RAW chain.
    for (int k = 0; k < HIDDEN; k += 8) {
        v2f a0, b0, a1, b1;
        a0.x = aptr[k];     a0.y = aptr[k + 1];
        b0.x = bptr[k] * msk;     b0.y = bptr[k + 1] * msk;
        a1.x = aptr[k + 4]; a1.y = aptr[k + 5];
        b1.x = bptr[k + 4] * msk; b1.y = bptr[k + 5] * msk;

        acc0 = __builtin_amdgcn_wmma_f32_16x16x4_f32(
            false, a0, false, b0, (short)0, acc0, false, false);
        acc1 = __builtin_amdgcn_wmma_f32_16x16x4_f32(
            false, a1, false, b1, (short)0, acc1, false, false);
    }

    // C/D layout: VGPR v -> M = v (lanes 0-15) / 8+v (lanes 16-31), N = lane&15
    const int baseM = mtile * 16 + ((lane >> 4) << 3);
    #pragma unroll
    for (int v = 0; v < 8; ++v) {
        P[(baseM + v) * P_STRIDE + ncol] = acc0[v] + acc1[v];
    }
}

// -----------------------------------------------------------------------------
// Stage 2: faithful first-match span lookup (argmax-of-equality semantics,
// default 0 on no match) + gather-add of precomputed partial logits.
// -----------------------------------------------------------------------------
__global__ __launch_bounds__(256) void gather_logits(
    const int*   __restrict__ spans,  // [512*2]
    const int*   __restrict__ ao,     // [100000*2*2]
    const float* __restrict__ P,      // [512*32]
    const float* __restrict__ bias,   // [13]
    float*       __restrict__ out)    // [100000*13]
{
    __shared__ int2  s_spans[NUM_SPANS];
    __shared__ float s_bias[16];

    const int t = threadIdx.x;
    for (int i = t; i < NUM_SPANS; i += blockDim.x)
        s_spans[i] = ((const int2*)spans)[i];
    if (t < NUM_CATS) s_bias[t] = bias[t];
    __syncthreads();

    const int p = blockIdx.x * blockDim.x + t;
    if (p >= NUM_PAIRS) return;

    const int a0 = ao[p * 4 + 0], a1 = ao[p * 4 + 1];
    const int o0 = ao[p * 4 + 2], o1 = ao[p * 4 + 3];

    int idx0 = 0, idx1 = 0;
    bool f0 = false, f1 = false;
    for (int k = 0; k < NUM_SPANS; ++k) {
        const int2 sp = s_spans[k];
        if (!f0 && sp.x == a0 && sp.y == a1) { idx0 = k; f0 = true; }
        if (!f1 && sp.x == o0 && sp.y == o1) { idx1 = k; f1 = true; }
        if (f0 && f1) break;
    }

    const float* __restrict__ pa = P + idx0 * P_STRIDE;            // aspect cols 0..12
    const float* __restrict__ po = P + idx1 * P_STRIDE + NUM_CATS; // opinion cols 13..25

    #pragma unroll
    for (int c = 0; c < NUM_CATS; ++c)
        out[p * NUM_CATS + c] = pa[c] + po[c] + s_bias[c];
}

extern "C" void kernel_launch(void* const* d_in, const int* in_sizes, int n_in,
                              void* d_out, int out_size, void* d_ws, size_t ws_size,
                              hipStream_t stream) {
    const float* emb   = (const float*)d_in[0];  // (1,512,768) f32
    const int*   spans = (const int*)  d_in[1];  // (512,2) i32
    const int*   ao    = (const int*)  d_in[2];  // (100000,2,2) i32
    const float* W     = (const float*)d_in[3];  // (13,1536) f32
    const float* b     = (const float*)d_in[4];  // (13,) f32
    float*       out   = (float*)d_out;          // (100000,13) f32
    float*       P     = (float*)d_ws;           // 512*32 f32 = 64 KB scratch

    partial_logits_wmma<<<64, 32, 0, stream>>>(emb, W, P);
    gather_logits<<<(NUM_PAIRS + 255) / 256, 256, 0, stream>>>(spans, ao, P, b, out);
}